// DynamicFP8Quantizer_66958540144942
// MI455X (gfx1250) — compile-verified
//
#include <hip/hip_runtime.h>
#include <stdint.h>

// ---------------------------------------------------------------------------
// Address-space helpers for CDNA5 async global->LDS builtins
// ---------------------------------------------------------------------------
#define AS1 __attribute__((address_space(1)))
#define AS3 __attribute__((address_space(3)))

typedef int vi4 __attribute__((vector_size(4 * sizeof(int))));

__device__ __forceinline__ void async_copy16(const float* gsrc, float* ldst) {
#if __has_builtin(__builtin_amdgcn_global_load_async_to_lds_b128)
  __builtin_amdgcn_global_load_async_to_lds_b128((AS1 vi4*)gsrc, (AS3 vi4*)ldst,
                                                 /*offset=*/0, /*cpol=*/0);
#else
  unsigned laddr = (unsigned)(size_t)(AS3 void*)ldst;
  asm volatile("global_load_async_to_lds_b128 %0, %1, off"
               :: "v"(laddr), "v"(gsrc) : "memory");
#endif
}

#if __has_builtin(__builtin_amdgcn_s_wait_asynccnt)
#define WAIT_ASYNC(n) __builtin_amdgcn_s_wait_asynccnt(n)
#else
#define WAIT_ASYNC(n) asm volatile("s_wait_asynccnt %0" :: "i"(n) : "memory")
#endif

// ---------------------------------------------------------------------------
// Workspace layout (u32 units). Total ~918 KB.
// ---------------------------------------------------------------------------
static constexpr unsigned kH1Off   = 0;                   // coarse hist: high 16 bits of |g|
static constexpr unsigned kH1N     = 32768;               // sign bit is 0 -> bins < 32768
static constexpr unsigned kMetaOff = kH1Off + kH1N;       // [0..2]=bin_j  [3..5]=cumBefore_j
static constexpr unsigned kMetaN   = 8;
static constexpr unsigned kH2Off   = kMetaOff + kMetaN;   // refine hists: 3 x 65536 (low 16 bits)
static constexpr unsigned kH2N     = 3u * 65536u;
static constexpr unsigned kQOff    = kH2Off + kH2N;       // q1,q2,q3 as float bits
static constexpr unsigned kQN      = 4;
static constexpr unsigned kWsU32   = kQOff + kQN;

// ---------------------------------------------------------------------------
// Per-region FP quantizer (regions use exp_bits 2,3,5,6 / mant bits 5,4,2,1)
// Exact bit-level version of the reference math (all intermediates normal).
// ---------------------------------------------------------------------------
__device__ __forceinline__ float quant_one(float x, float g, float q1, float q2, float q3) {
  const float ga = fabsf(g);
  const int r = (ga > q1) + (ga > q2) + (ga > q3);

  const float maxv   = r == 0 ? 7.75f    : r == 1 ? 31.0f    : r == 2 ? 114688.0f : 6442450944.0f;
  const float minv   = r == 0 ? 0x1p-5f  : r == 1 ? 0x1p-6f  : r == 2 ? 0x1p-16f  : 0x1p-31f;
  const int   minE   = r == 0 ? 0        : r == 1 ? -2       : r == 2 ? -14       : -30;
  const int   maxE   = r == 0 ? 2        : r == 1 ? 4        : r == 2 ? 16        : 32;
  const float levels = r == 0 ? 32.0f    : r == 1 ? 16.0f    : r == 2 ? 4.0f      : 2.0f;
  const float invlev = r == 0 ? 0.03125f : r == 1 ? 0.0625f  : r == 2 ? 0.25f     : 0.5f;

  const float xc = fminf(fmaxf(x, -maxv), maxv);
  const unsigned sbit = __float_as_uint(xc) & 0x80000000u;
  float xa = fabsf(xc);
  const bool zero = xa < minv;
  xa = fmaxf(xa, minv);
  int e = (int)(__float_as_uint(xa) >> 23) - 127;   // exact floor(log2(xa)); xa is normal
  e = e < minE ? minE : e;
  e = e > maxE ? maxE : e;
  const float p2  = __uint_as_float((unsigned)(e + 127) << 23);
  const float p2i = __uint_as_float((unsigned)(127 - e) << 23);
  const float mf  = __builtin_fmaf(xa, p2i, -1.0f);
  float mq = rintf(mf * levels) * invlev;             // round half-to-even == jnp.round
  mq = fminf(fmaxf(mq, 0.0f), 1.0f - invlev);
  float res = __uint_as_float(__float_as_uint(p2 * (1.0f + mq)) | sbit);
  return zero ? 0.0f : res;
}

// ---------------------------------------------------------------------------
// Kernel 0: zero workspace
// ---------------------------------------------------------------------------
__global__ void zero_ws_kernel(unsigned* __restrict__ ws, unsigned count) {
  const unsigned stride = gridDim.x * blockDim.x;
  for (unsigned i = blockIdx.x * blockDim.x + threadIdx.x; i < count; i += stride)
    ws[i] = 0u;
}

// ---------------------------------------------------------------------------
// Kernel 1: coarse histogram of high 16 bits of |g|, LDS-privatized.
// 32768 bins * 4B = 128 KB LDS -- only possible with CDNA5's 320 KB LDS.
// ---------------------------------------------------------------------------
__global__ __launch_bounds__(1024) void hist1_kernel(const float* __restrict__ g,
                                                     unsigned n4, unsigned nTail,
                                                     unsigned* __restrict__ gh) {
  __shared__ unsigned lh[32768];
  for (unsigned i = threadIdx.x; i < 32768u; i += 1024u) lh[i] = 0u;
  __syncthreads();

  const size_t stride = (size_t)gridDim.x * 1024u;
  const float4* g4 = (const float4*)g;
  for (size_t i = (size_t)blockIdx.x * 1024u + threadIdx.x; i < n4; i += stride) {
    float4 v = g4[i];
    atomicAdd(&lh[(__float_as_uint(v.x) & 0x7FFFFFFFu) >> 16], 1u);
    atomicAdd(&lh[(__float_as_uint(v.y) & 0x7FFFFFFFu) >> 16], 1u);
    atomicAdd(&lh[(__float_as_uint(v.z) & 0x7FFFFFFFu) >> 16], 1u);
    atomicAdd(&lh[(__float_as_uint(v.w) & 0x7FFFFFFFu) >> 16], 1u);
  }
  if (blockIdx.x == 0 && threadIdx.x < nTail) {
    float v = g[(size_t)n4 * 4u + threadIdx.x];
    atomicAdd(&lh[(__float_as_uint(v) & 0x7FFFFFFFu) >> 16], 1u);
  }
  __syncthreads();
  for (unsigned i = threadIdx.x; i < 32768u; i += 1024u) {
    unsigned c = lh[i];
    if (c) atomicAdd(&gh[i], c);
  }
}

// ---------------------------------------------------------------------------
// Kernel 2: pick coarse bin + cumulative-before for each rank (one block).
// ---------------------------------------------------------------------------
__global__ __launch_bounds__(256) void select_bins_kernel(const unsigned* __restrict__ gh,
                                                          unsigned r1, unsigned r2, unsigned r3,
                                                          unsigned* __restrict__ meta) {
  __shared__ unsigned psum[256];
  const unsigned t = threadIdx.x;
  unsigned s = 0;
  for (unsigned i = 0; i < 128u; ++i) s += gh[t * 128u + i];
  psum[t] = s;
  __syncthreads();
  if (t == 0) {
    unsigned run = 0;
    for (int i = 0; i < 256; ++i) { unsigned v = psum[i]; psum[i] = run; run += v; }
  }
  __syncthreads();
  unsigned cum = psum[t];
  const unsigned ranks[3] = {r1, r2, r3};
  for (unsigned i = 0; i < 128u; ++i) {
    const unsigned c = gh[t * 128u + i];
    const unsigned nc = cum + c;
    #pragma unroll
    for (int j = 0; j < 3; ++j) {
      if (ranks[j] >= cum && ranks[j] < nc) {  // order statistic falls in this bin
        meta[j] = t * 128u + i;
        meta[3 + j] = cum;
      }
    }
    cum = nc;
  }
}

// ---------------------------------------------------------------------------
// Kernel 3: refine histogram of low 16 bits, only for the 3 candidate bins.
// ---------------------------------------------------------------------------
__global__ __launch_bounds__(256) void hist2_kernel(const float* __restrict__ g,
                                                    unsigned n4, unsigned nTail,
                                                    const unsigned* __restrict__ meta,
                                                    unsigned* __restrict__ h2) {
  const unsigned b0 = meta[0], b1 = meta[1], b2 = meta[2];
  const size_t stride = (size_t)gridDim.x * blockDim.x;
  const float4* g4 = (const float4*)g;
  for (size_t i = (size_t)blockIdx.x * blockDim.x + threadIdx.x; i < n4; i += stride) {
    float4 v = g4[i];
    const float vv[4] = {v.x, v.y, v.z, v.w};
    #pragma unroll
    for (int k = 0; k < 4; ++k) {
      const unsigned bits = __float_as_uint(vv[k]) & 0x7FFFFFFFu;
      const unsigned hi = bits >> 16, lo = bits & 0xFFFFu;
      if (hi == b0) atomicAdd(&h2[lo], 1u);
      if (hi == b1) atomicAdd(&h2[65536u + lo], 1u);
      if (hi == b2) atomicAdd(&h2[131072u + lo], 1u);
    }
  }
  if (blockIdx.x == 0 && threadIdx.x < nTail) {
    const unsigned bits = __float_as_uint(g[(size_t)n4 * 4u + threadIdx.x]) & 0x7FFFFFFFu;
    const unsigned hi = bits >> 16, lo = bits & 0xFFFFu;
    if (hi == b0) atomicAdd(&h2[lo], 1u);
    if (hi == b1) atomicAdd(&h2[65536u + lo], 1u);
    if (hi == b2) atomicAdd(&h2[131072u + lo], 1u);
  }
}

// ---------------------------------------------------------------------------
// Kernel 4: finalize exact order-statistic thresholds (3 blocks, one per q).
// ---------------------------------------------------------------------------
__global__ __launch_bounds__(256) void final_q_kernel(const unsigned* __restrict__ h2,
                                                      const unsigned* __restrict__ meta,
                                                      unsigned r1, unsigned r2, unsigned r3,
                                                      float* __restrict__ qout) {
  const unsigned j = blockIdx.x;
  const unsigned* h = h2 + (size_t)j * 65536u;
  const unsigned rank = (j == 0 ? r1 : j == 1 ? r2 : r3) - meta[3 + j];  // rank within bin
  __shared__ unsigned psum[256];
  const unsigned t = threadIdx.x;
  unsigned s = 0;
  for (unsigned i = 0; i < 256u; ++i) s += h[t * 256u + i];
  psum[t] = s;
  __syncthreads();
  if (t == 0) {
    unsigned run = 0;
    for (int i = 0; i < 256; ++i) { unsigned v = psum[i]; psum[i] = run; run += v; }
  }
  __syncthreads();
  unsigned cum = psum[t];
  for (unsigned i = 0; i < 256u; ++i) {
    const unsigned c = h[t * 256u + i];
    const unsigned nc = cum + c;
    if (rank >= cum && rank < nc) {
      const unsigned low = t * 256u + i;
      qout[j] = __uint_as_float((meta[j] << 16) | low);
    }
    cum = nc;
  }
}

// ---------------------------------------------------------------------------
// Kernel 5: main streaming quantizer. Double-buffered async global->LDS for
// both x and g; trip count is grid-uniform so per-wave ASYNCcnt waits are
// exact (async tile i complete once <=2 newer ops remain outstanding).
// ---------------------------------------------------------------------------
__global__ __launch_bounds__(256) void quantize_kernel(const float* __restrict__ x,
                                                       const float* __restrict__ g,
                                                       float* __restrict__ out,
                                                       const float* __restrict__ qv,
                                                       unsigned n4, unsigned nTail) {
  __shared__ __align__(16) float sx[2][256 * 4];
  __shared__ __align__(16) float sg[2][256 * 4];
  const float q1 = qv[0], q2 = qv[1], q3 = qv[2];
  const unsigned tid = threadIdx.x;
  const unsigned stride4 = gridDim.x * 256u;
  const unsigned base4 = blockIdx.x * 256u + tid;
  const unsigned R = n4 / stride4;  // identical for every thread in the grid

  if (R > 0) {
    async_copy16(x + 4ull * base4, &sx[0][tid * 4u]);
    async_copy16(g + 4ull * base4, &sg[0][tid * 4u]);
  }
  unsigned buf = 0;
  for (unsigned r = 0; r < R; ++r) {
    const unsigned i4 = r * stride4 + base4;
    if (r + 1 < R) {
      const unsigned p4 = i4 + stride4;
      async_copy16(x + 4ull * p4, &sx[buf ^ 1u][tid * 4u]);
      async_copy16(g + 4ull * p4, &sg[buf ^ 1u][tid * 4u]);
      WAIT_ASYNC(2);  // tile r's two async ops have retired
    } else {
      WAIT_ASYNC(0);
    }
    const float4 xv = *(const float4*)&sx[buf][tid * 4u];
    const float4 gv = *(const float4*)&sg[buf][tid * 4u];
    float4 ov;
    ov.x = quant_one(xv.x, gv.x, q1, q2, q3);
    ov.y = quant_one(xv.y, gv.y, q1, q2, q3);
    ov.z = quant_one(xv.z, gv.z, q1, q2, q3);
    ov.w = quant_one(xv.w, gv.w, q1, q2, q3);
    *(float4*)(out + 4ull * i4) = ov;
    buf ^= 1u;
  }
  // remaining float4s (non-uniform): plain vector loads
  const unsigned t4 = R * stride4 + base4;
  if (t4 < n4) {
    const float4 xv = *(const float4*)(x + 4ull * t4);
    const float4 gv = *(const float4*)(g + 4ull * t4);
    float4 ov;
    ov.x = quant_one(xv.x, gv.x, q1, q2, q3);
    ov.y = quant_one(xv.y, gv.y, q1, q2, q3);
    ov.z = quant_one(xv.z, gv.z, q1, q2, q3);
    ov.w = quant_one(xv.w, gv.w, q1, q2, q3);
    *(float4*)(out + 4ull * t4) = ov;
  }
  // scalar tail (n % 4)
  if (blockIdx.x == 0 && tid < nTail) {
    const size_t i = 4ull * n4 + tid;
    out[i] = quant_one(x[i], g[i], q1, q2, q3);
  }
}

// ---------------------------------------------------------------------------
// Host launcher
// ---------------------------------------------------------------------------
extern "C" void kernel_launch(void* const* d_in, const int* in_sizes, int n_in,
                              void* d_out, int out_size, void* d_ws, size_t ws_size,
                              hipStream_t stream) {
  (void)n_in; (void)out_size; (void)ws_size;
  const float* x = (const float*)d_in[0];
  const float* g = (const float*)d_in[1];
  float* out = (float*)d_out;
  unsigned* ws = (unsigned*)d_ws;

  const size_t n = (size_t)in_sizes[0];
  const unsigned n4 = (unsigned)(n / 4u);
  const unsigned nTail = (unsigned)(n & 3u);

  // order-statistic ranks k = floor(p * (n-1))  (classifies identically to
  // the interpolated quantile: no sample lies strictly between order stats)
  const unsigned r1 = (unsigned)((n - 1) / 4);
  const unsigned r2 = (unsigned)((n - 1) / 2);
  const unsigned r3 = (unsigned)((3ull * (n - 1)) / 4);

  zero_ws_kernel<<<512, 256, 0, stream>>>(ws, kWsU32);
  hist1_kernel<<<1024, 1024, 0, stream>>>(g, n4, nTail, ws + kH1Off);
  select_bins_kernel<<<1, 256, 0, stream>>>(ws + kH1Off, r1, r2, r3, ws + kMetaOff);
  hist2_kernel<<<2048, 256, 0, stream>>>(g, n4, nTail, ws + kMetaOff, ws + kH2Off);
  final_q_kernel<<<3, 256, 0, stream>>>(ws + kH2Off, ws + kMetaOff, r1, r2, r3,
                                        (float*)(ws + kQOff));
  quantize_kernel<<<4096, 256, 0, stream>>>(x, g, out, (const float*)(ws + kQOff),
                                            n4, nTail);
}